// BottomUpAttention_67190468378729
// MI455X (gfx1250) — compile-verified
//
#include <hip/hip_runtime.h>
#include <math.h>

#define Bsz  4
#define Nn   300
#define TOPk 36
#define Cc   2048
#define Dd   2048
#define CONF 0.3f
#define IOU  0.7f

typedef __attribute__((ext_vector_type(2))) float v2f;
typedef __attribute__((ext_vector_type(4))) float v4f;
typedef __attribute__((ext_vector_type(8))) float v8f;

// ---------------------------------------------------------------------------
// Kernel 1: per-image confidence filter + greedy NMS + top-36 + level/key rank
// ---------------------------------------------------------------------------
__global__ __launch_bounds__(256) void k_select(
    const float* __restrict__ boxes, const float* __restrict__ scores,
    float* __restrict__ roi_box, int* __restrict__ roi_level,
    int* __restrict__ roi_dest, int* __restrict__ rowvalid)
{
  __shared__ float sc[Nn];
  __shared__ float bx[Nn][4];
  __shared__ int   ord[Nn];
  __shared__ int   keep[Nn];
  __shared__ float sb[Nn][4];
  __shared__ float area[Nn];
  __shared__ int   take[TOPk];
  __shared__ int   keys[TOPk];
  __shared__ int   nval;

  const int b = blockIdx.x;
  const int tid = threadIdx.x;

  for (int i = tid; i < Nn; i += 256) {
    sc[i] = scores[b * Nn + i];
    for (int k = 0; k < 4; k++) bx[i][k] = boxes[(b * Nn + i) * 4 + k];
  }
  __syncthreads();

  // stable descending argsort via rank counting
  for (int i = tid; i < Nn; i += 256) {
    float si = sc[i];
    int r = 0;
    for (int j = 0; j < Nn; j++) {
      float sj = sc[j];
      r += (sj > si) || (sj == si && j < i);
    }
    ord[r] = i;
  }
  __syncthreads();

  for (int r = tid; r < Nn; r += 256) {
    int i = ord[r];
    float x1 = bx[i][0], y1 = bx[i][1], x2 = bx[i][2], y2 = bx[i][3];
    sb[r][0] = x1; sb[r][1] = y1; sb[r][2] = x2; sb[r][3] = y2;
    area[r] = (x2 - x1) * (y2 - y1);
    keep[r] = sc[i] > CONF ? 1 : 0;
  }
  __syncthreads();

  // greedy NMS (serial over i, parallel over j)
  for (int i = 0; i < Nn; i++) {
    int ki = keep[i];
    if (ki) {
      float ax1 = sb[i][0], ay1 = sb[i][1], ax2 = sb[i][2], ay2 = sb[i][3];
      float ai = area[i];
      for (int j = i + 1 + tid; j < Nn; j += 256) {
        float xx1 = fmaxf(ax1, sb[j][0]);
        float yy1 = fmaxf(ay1, sb[j][1]);
        float xx2 = fminf(ax2, sb[j][2]);
        float yy2 = fminf(ay2, sb[j][3]);
        float inter = fmaxf(xx2 - xx1, 0.f) * fmaxf(yy2 - yy1, 0.f);
        float iou = inter / (ai + area[j] - inter);
        if (iou > IOU) keep[j] = 0;
      }
    }
    __syncthreads();
  }

  if (tid == 0) {
    int cnt = 0;
    for (int r = 0; r < Nn && cnt < TOPk; r++)
      if (keep[r]) take[cnt++] = r;
    nval = cnt;
    for (int t = cnt; t < TOPk; t++) take[t] = 0;  // dummy; rows masked later
  }
  __syncthreads();

  if (tid < TOPk) {
    int t = tid;
    int valid = t < nval;
    int r = take[t];
    float x1 = sb[r][0], y1 = sb[r][1], x2 = sb[r][2], y2 = sb[r][3];
    float dx = x2 - x1, dy = y2 - y1;
    float diag = sqrtf(dx * dx + dy * dy);
    float lv = floorf(4.0f + log2f(diag * (4.0f / 224.0f)));
    lv = fminf(fmaxf(lv, 2.0f), 5.0f);
    int lvl = (int)lv - 2;
    keys[t] = valid ? (lvl * TOPk + t) : (4 * TOPk + t);
    int bt = b * TOPk + t;
    roi_box[bt * 4 + 0] = x1; roi_box[bt * 4 + 1] = y1;
    roi_box[bt * 4 + 2] = x2; roi_box[bt * 4 + 3] = y2;
    roi_level[bt] = lvl;
  }
  __syncthreads();

  if (tid < TOPk) {
    int t = tid;
    int kk = keys[t];
    int rank = 0;  // keys are unique -> rank = final position (stable argsort)
    for (int u = 0; u < TOPk; u++) rank += keys[u] < kk;
    int bt = b * TOPk + t;
    roi_dest[bt] = rank;
    rowvalid[b * TOPk + rank] = (t < nval) ? 1 : 0;
  }
}

// ---------------------------------------------------------------------------
// Kernel 2: ROIAlign(7x7, sr=2) + global mean over 14x14 samples.
// Writes pooled row at its FINAL permuted row index so the GEMM is dense.
// ---------------------------------------------------------------------------
__global__ __launch_bounds__(256) void k_pool(
    const float* __restrict__ f0, const float* __restrict__ f1,
    const float* __restrict__ f2, const float* __restrict__ f3,
    const float* __restrict__ roi_box, const int* __restrict__ roi_level,
    const int* __restrict__ roi_dest, const int* __restrict__ rowvalid,
    float* __restrict__ P)
{
  const int bt = blockIdx.x;
  const int b = bt / TOPk;
  const int tid = threadIdx.x;
  const int dest = b * TOPk + roi_dest[bt];
  float* dst = P + (size_t)dest * Cc;

  if (!rowvalid[dest]) {  // uniform per block
    for (int c = tid; c < Cc; c += 256) dst[c] = 0.f;
    return;
  }

  int lvl = roi_level[bt];
  const float* fm; int Hs;
  switch (lvl) {
    case 0:  fm = f0; Hs = 100; break;
    case 1:  fm = f1; Hs = 50;  break;
    case 2:  fm = f2; Hs = 25;  break;
    default: fm = f3; Hs = 13;  break;
  }
  const int HW = Hs * Hs;
  const float* base = fm + (size_t)b * Cc * HW;

  __shared__ int   y0s[14], y1s[14], x0s[14], x1s[14];
  __shared__ float wys[14], wxs[14];
  __shared__ int   youts[14], xouts[14];

  float x1b = roi_box[bt * 4 + 0], y1b = roi_box[bt * 4 + 1];
  float x2b = roi_box[bt * 4 + 2], y2b = roi_box[bt * 4 + 3];
  float rw = fmaxf(x2b - x1b, 1.f), rh = fmaxf(y2b - y1b, 1.f);

  if (tid < 14) {
    int i = tid;
    float ys = y1b + (float)(i / 2) * (rh / 7.f) + ((float)(i % 2) + 0.5f) * (rh / 14.f);
    youts[i] = (ys < -1.f) || (ys > (float)Hs);
    float y = fminf(fmaxf(ys, 0.f), (float)(Hs - 1));
    int y0 = (int)floorf(y);
    y0s[i] = y0;
    y1s[i] = min(y0 + 1, Hs - 1);
    wys[i] = y - (float)y0;

    float xs = x1b + (float)(i / 2) * (rw / 7.f) + ((float)(i % 2) + 0.5f) * (rw / 14.f);
    xouts[i] = (xs < -1.f) || (xs > (float)Hs);
    float x = fminf(fmaxf(xs, 0.f), (float)(Hs - 1));
    int x0 = (int)floorf(x);
    x0s[i] = x0;
    x1s[i] = min(x0 + 1, Hs - 1);
    wxs[i] = x - (float)x0;
  }
  __syncthreads();

  float acc[8];
  #pragma unroll
  for (int cc = 0; cc < 8; cc++) acc[cc] = 0.f;

  for (int sy = 0; sy < 14; sy++) {
    if (youts[sy]) continue;  // whole row of samples contributes zero
    int y0 = y0s[sy], y1 = y1s[sy];
    float wy = wys[sy];
    for (int sx = 0; sx < 14; sx++) {
      if (xouts[sx]) continue;  // sample contributes zero
      int x0 = x0s[sx], x1 = x1s[sx];
      float wx = wxs[sx];
      float w00 = (1.f - wy) * (1.f - wx), w01 = (1.f - wy) * wx;
      float w10 = wy * (1.f - wx),          w11 = wy * wx;
      int o00 = y0 * Hs + x0, o01 = y0 * Hs + x1;
      int o10 = y1 * Hs + x0, o11 = y1 * Hs + x1;
      #pragma unroll
      for (int cc = 0; cc < 8; cc++) {
        const float* p = base + (size_t)(tid + cc * 256) * HW;
        acc[cc] += p[o00] * w00 + p[o01] * w01 + p[o10] * w10 + p[o11] * w11;
      }
    }
  }
  const float inv = 1.0f / 196.f;
  #pragma unroll
  for (int cc = 0; cc < 8; cc++) dst[tid + cc * 256] = acc[cc] * inv;
}

// ---------------------------------------------------------------------------
// Kernel 3: OUT = P @ W^T + b, masked by rowvalid. f32 WMMA 16x16x4.
// Block = 8 waves, tile 16(M) x 128(N); K staged in 64-wide LDS chunks.
// ---------------------------------------------------------------------------
__global__ __launch_bounds__(256) void k_gemm(
    const float* __restrict__ P, const float* __restrict__ Wm,
    const float* __restrict__ bias, const int* __restrict__ rowvalid,
    float* __restrict__ out)
{
  __shared__ __align__(16) float As[16][64];    // 4 KB
  __shared__ __align__(16) float Bs[128][64];   // 32 KB

  const int tid  = threadIdx.x;
  const int lane = tid & 31;
  const int wv   = tid >> 5;                 // wave 0..7 -> N subtile
  const int mtile = blockIdx.y;              // 0..8
  const int nbase = blockIdx.x * 128;        // 0..15 * 128

  const int arow = tid >> 4;                 // 0..15
  const int ac4  = (tid & 15) << 2;          // 0,4,...,60

  const int m  = lane & 15;                  // A-frag row
  const int kp = lane >> 4;                  // K pair select (0 -> K0/K1, 1 -> K2/K3)
  const int nl = (wv << 4) + (lane & 15);    // B row within 128-wide tile

  v8f acc = {};

  for (int k0 = 0; k0 < Cc; k0 += 64) {
    // hint next W chunk toward L2 while we compute on this one
    if (k0 + 64 < Cc)
      __builtin_prefetch(&Wm[(size_t)(nbase + (tid & 127)) * Cc + k0 + 64], 0, 1);

    __syncthreads();
    // stage A: P[mtile*16 .. +16)[k0 .. k0+64)
    {
      const float* src = P + (size_t)(mtile * 16 + arow) * Cc + k0 + ac4;
      *(v4f*)&As[arow][ac4] = *(const v4f*)src;
    }
    // stage B: W[nbase .. nbase+128)[k0 .. k0+64)
    #pragma unroll
    for (int i = 0; i < 8; i++) {
      int idx = tid + i * 256;
      int br  = idx >> 4;
      int bc4 = (idx & 15) << 2;
      const float* src = Wm + (size_t)(nbase + br) * Cc + k0 + bc4;
      *(v4f*)&Bs[br][bc4] = *(const v4f*)src;
    }
    __syncthreads();

    #pragma unroll
    for (int ks = 0; ks < 64; ks += 4) {
      // ISA 7.12.2 layouts: VGPR0 = K{0|2}, VGPR1 = K{1|3} across half-waves
      v2f a = *(const v2f*)&As[m][ks + 2 * kp];
      v2f bf = *(const v2f*)&Bs[nl][ks + 2 * kp];
      acc = __builtin_amdgcn_wmma_f32_16x16x4_f32(
          /*neg_a=*/false, a, /*neg_b=*/false, bf,
          /*c_mod=*/(short)0, acc, /*reuse_a=*/false, /*reuse_b=*/false);
    }
  }

  // epilogue: D vgpr i -> row (i + 8*half), col (lane % 16)
  const int col  = lane & 15;
  const int half = lane >> 4;
  const int n = nbase + (wv << 4) + col;
  const float bv = bias[n];
  #pragma unroll
  for (int i = 0; i < 8; i++) {
    int grow = mtile * 16 + i + 8 * half;
    float v = rowvalid[grow] ? (acc[i] + bv) : 0.f;
    out[(size_t)grow * Dd + n] = v;
  }
}

// ---------------------------------------------------------------------------
extern "C" void kernel_launch(void* const* d_in, const int* in_sizes, int n_in,
                              void* d_out, int out_size, void* d_ws, size_t ws_size,
                              hipStream_t stream) {
  const float* boxes  = (const float*)d_in[0];
  const float* scores = (const float*)d_in[1];
  const float* f0     = (const float*)d_in[2];
  const float* f1     = (const float*)d_in[3];
  const float* f2     = (const float*)d_in[4];
  const float* f3     = (const float*)d_in[5];
  const float* Wm     = (const float*)d_in[6];
  const float* bv     = (const float*)d_in[7];
  float* out = (float*)d_out;

  // workspace layout
  float* P        = (float*)d_ws;                       // 144*2048 f32
  int*   rowvalid = (int*)(P + (size_t)Bsz * TOPk * Cc);// 144
  int*   roi_level = rowvalid + Bsz * TOPk;             // 144
  int*   roi_dest  = roi_level + Bsz * TOPk;            // 144
  float* roi_box   = (float*)(roi_dest + Bsz * TOPk);   // 144*4

  k_select<<<Bsz, 256, 0, stream>>>(boxes, scores, roi_box, roi_level,
                                    roi_dest, rowvalid);
  k_pool<<<Bsz * TOPk, 256, 0, stream>>>(f0, f1, f2, f3, roi_box, roi_level,
                                         roi_dest, rowvalid, P);
  k_gemm<<<dim3(16, 9), 256, 0, stream>>>(P, Wm, bv, rowvalid, out);
}